// Codebook_76897094468462
// MI455X (gfx1250) — compile-verified
//
#include <hip/hip_runtime.h>

// VQ-VAE codebook quantization for MI455X (gfx1250, wave32).
// B=32, C=256, H=W=32, K=1024. Faithful to the torch reshape bug:
//   ZZ[b][i][n] = z[b][n&255][ i*4 + (n>>8) ]   (plane index over h*w)
// score[k][n] = ||e_k||^2 - 2 * sum_i E[k][i] * ZZ[b][i][n]; idx = argmin_k.
// out = ( E[idx] scattered to [b,c,h,w],  idx as float,  1.25*mean((zq-zp)^2) ).

typedef __attribute__((ext_vector_type(2))) float v2f;
typedef __attribute__((ext_vector_type(8))) float v8f;

#define A_STRIDE 260   // 16 rows x 260 floats: conflict-free ds_load_b64 frags

// ---------------- kernel 0: enorm[k] = sum_c E[k][c]^2 -------------------
__global__ __launch_bounds__(256) void vq_enorm_kernel(
    const float* __restrict__ E, float* __restrict__ enorm)
{
    const int lane = threadIdx.x & 31;
    const int k = blockIdx.x * 8 + (threadIdx.x >> 5);   // 128 blocks * 8 waves
    const float* row = E + (size_t)k * 256;
    float s = 0.f;
#pragma unroll
    for (int u = 0; u < 8; ++u) {
        float v = row[lane + 32 * u];
        s += v * v;
    }
#pragma unroll
    for (int m = 16; m >= 1; m >>= 1) s += __shfl_xor(s, m, 32);
    if (lane == 0) enorm[k] = s;
}

// ---------------- kernel 1: fused GEMM (fp32 WMMA) + argmin --------------
// grid = 32 b * 8 n-blocks = 256 WGs; 256 threads = 8 waves.
// wave w handles n-tile n0 = nblk*128 + w*16 (16 columns).
__global__ __launch_bounds__(256) void vq_argmin_kernel(
    const float* __restrict__ z, const float* __restrict__ E,
    const float* __restrict__ enorm, int* __restrict__ idxbuf)
{
    __shared__ float sA[16 * A_STRIDE];   // embedding k-tile, padded
    __shared__ float sEn[1024];           // ||e_k||^2

    const int tid  = threadIdx.x;
    const int lane = tid & 31;
    const int nn   = lane & 15;           // N within tile
    const int h    = lane >> 4;           // lane-half (K/M split)
    const int wid  = tid >> 5;            // wave 0..7
    const int b    = blockIdx.x >> 3;
    const int nblk = blockIdx.x & 7;
    const int n0   = nblk * 128 + wid * 16;
    const int n    = n0 + nn;

    // stage enorm -> LDS (256 threads * float4 = 4KB)
    ((float4*)sEn)[tid] = ((const float4*)enorm)[tid];

    // ---- load ALL 64 B-fragments into registers (128 VGPRs/lane) ----
    // B frag step s (i = 4s): lane needs ZZ[i+2h][n], ZZ[i+2h+1][n]
    // ZZ[i'][n] = plane[i'*4], plane = z[b][n&255][.] + (n>>8)
    const float* plane =
        z + (((size_t)b * 256 + (size_t)(n & 255)) * 1024) + (n >> 8);
    v2f bfr[64];
#pragma unroll
    for (int s = 0; s < 64; ++s) {
        const int x = 16 * s + 8 * h;
        v2f f;
        f.x = plane[x];
        f.y = plane[x + 4];
        bfr[s] = f;
    }

    float best = 3.4e38f;
    int   bidx = 0;

    for (int kt = 0; kt < 64; ++kt) {
        const int k0 = kt * 16;
        __syncthreads();   // previous A tile fully consumed (also covers sEn)
        {   // cooperative stage: E[k0+r][0..255] -> sA[r][0..255], 16B vectors
            const int r  = tid >> 4;
            const int c0 = (tid & 15) * 16;
            const float4* src = (const float4*)(E + (size_t)(k0 + r) * 256 + c0);
            float4* dst = (float4*)(sA + r * A_STRIDE + c0);
            dst[0] = src[0]; dst[1] = src[1]; dst[2] = src[2]; dst[3] = src[3];
        }
        __syncthreads();

        v8f acc = {0.f, 0.f, 0.f, 0.f, 0.f, 0.f, 0.f, 0.f};
        const int arow = (lane & 15) * A_STRIDE + 2 * h;
#pragma unroll
        for (int s = 0; s < 64; ++s) {
            // A frag: E[k0 + (lane&15)][4s + 2h .. +1]  (ds_load_b64)
            v2f afrag = *reinterpret_cast<const v2f*>(&sA[arow + 4 * s]);
            acc = __builtin_amdgcn_wmma_f32_16x16x4_f32(
                false, afrag, false, bfr[s], (short)0, acc, false, false);
        }

        // fused argmin update: VGPR j, lane-half h holds k = k0 + 8h + j
        const int kb = k0 + 8 * h;
#pragma unroll
        for (int j = 0; j < 8; ++j) {
            float sc = sEn[kb + j] - 2.0f * acc[j];
            if (sc < best) { best = sc; bidx = kb + j; }
        }
    }

    // combine the two lane halves (they own k%16 in [0,8) and [8,16))
    {
        float ob = __shfl_xor(best, 16, 32);
        int   oi = __shfl_xor(bidx, 16, 32);
        if (ob < best || (ob == best && oi < bidx)) { best = ob; bidx = oi; }
    }
    if (h == 0) idxbuf[b * 1024 + n0 + nn] = bidx;
}

// ---------------- kernel 2: gather + idx emit + loss partials ------------
// grid = 32 b * 4 n-chunks = 128 WGs; thread -> one n, loop over c (coalesced).
__global__ __launch_bounds__(256) void vq_gather_loss_kernel(
    const float* __restrict__ z, const float* __restrict__ E,
    const int* __restrict__ idxbuf, float* __restrict__ out,
    float* __restrict__ partials)
{
    __shared__ float red[256];
    const int tid = threadIdx.x;
    const int b   = blockIdx.x >> 2;
    const int n   = (blockIdx.x & 3) * 256 + tid;

    const int kidx = idxbuf[b * 1024 + n];
    out[8388608 + b * 1024 + n] = (float)kidx;   // idx output (as float)

    const float* erow  = E + (size_t)kidx * 256;
    const float* zbase = z   + (size_t)b * 262144 + n;
    float*       obase = out + (size_t)b * 262144 + n;

    float acc = 0.f;
#pragma unroll 4
    for (int c = 0; c < 256; ++c) {
        float e  = erow[c];
        float zv = zbase[(size_t)c * 1024];
        float d  = e - zv;
        acc += d * d;
        obase[(size_t)c * 1024] = e;             // z_q_st in [b,c,h,w]
    }

    red[tid] = acc;
    __syncthreads();
    for (int off = 128; off > 0; off >>= 1) {
        if (tid < off) red[tid] += red[tid + off];
        __syncthreads();
    }
    if (tid == 0) partials[blockIdx.x] = red[0];
}

// ---------------- kernel 3: loss finalize --------------------------------
__global__ __launch_bounds__(128) void vq_loss_finalize(
    const float* __restrict__ partials, float* __restrict__ out)
{
    __shared__ float red[128];
    const int tid = threadIdx.x;
    red[tid] = partials[tid];
    __syncthreads();
    for (int off = 64; off > 0; off >>= 1) {
        if (tid < off) red[tid] += red[tid + off];
        __syncthreads();
    }
    // (1 + BETA) * mean over B*C*H*W = 1.25 / 8388608
    if (tid == 0) out[8421376] = red[0] * (1.25f / 8388608.0f);
}

// ---------------- launcher ----------------------------------------------
extern "C" void kernel_launch(void* const* d_in, const int* in_sizes, int n_in,
                              void* d_out, int out_size, void* d_ws, size_t ws_size,
                              hipStream_t stream)
{
    (void)in_sizes; (void)n_in; (void)out_size; (void)ws_size;
    const float* z = (const float*)d_in[0];   // [32,256,32,32]
    const float* E = (const float*)d_in[1];   // [1024,256]
    float* out = (float*)d_out;               // 8388608 + 32768 + 1 floats

    // workspace: enorm (4KB) | idxbuf (128KB) | partials (512B)
    float* enorm    = (float*)d_ws;
    int*   idxbuf   = (int*)((char*)d_ws + 4096);
    float* partials = (float*)((char*)d_ws + 4096 + 32768 * 4);

    vq_enorm_kernel<<<128, 256, 0, stream>>>(E, enorm);
    vq_argmin_kernel<<<256, 256, 0, stream>>>(z, E, enorm, idxbuf);
    vq_gather_loss_kernel<<<128, 256, 0, stream>>>(z, E, idxbuf, out, partials);
    vq_loss_finalize<<<1, 128, 0, stream>>>(partials, out);
}